// QQ_58119497449488
// MI455X (gfx1250) — compile-verified
//
#include <hip/hip_runtime.h>
#include <cstdint>
#include <cstddef>

#define NDIM    6
#define JLEV    8
#define CHAN    64
#define ROWS    256
#define THREADS 256

typedef float v2f __attribute__((ext_vector_type(2)));
typedef float v4f __attribute__((ext_vector_type(4)));
typedef int   v4i __attribute__((ext_vector_type(4)));
typedef __attribute__((address_space(3))) v4i lds_v4i;

// gfx12-style CPol: TH in [2:0] (1 = NT), scope in [4:3] (0 = default)
#define CPOL_RT 0
#define CPOL_NT 1

#if defined(__AMDGCN__) && \
    __has_builtin(__builtin_amdgcn_global_load_async_to_lds_b128) && \
    __has_builtin(__builtin_amdgcn_global_store_async_from_lds_b128) && \
    __has_builtin(__builtin_amdgcn_s_wait_asynccnt)
#define USE_ASYNC 1
#else
#define USE_ASYNC 0
#endif

__global__ __launch_bounds__(THREADS) void fsq_gather_kernel(
    const float* __restrict__ z,
    const float* __restrict__ perc,      // [JLEV, NDIM], sorted along level axis
    const float* __restrict__ codebook,  // [262144, CHAN] = 64 MB, L2-resident
    float* __restrict__ out_q,           // [B, CHAN]
    float* __restrict__ out_big,         // [B] (float-encoded int, exact up to 2^24)
    int B)
{
    __shared__ float pshm[JLEV * NDIM];
    __shared__ int   idx_shm[ROWS];
    __shared__ __align__(16) float stage[ROWS * CHAN];   // 64 KB staging for async DMA

    const int t     = threadIdx.x;
    const int rbase = blockIdx.x * ROWS;

    if (t < JLEV * NDIM) pshm[t] = perc[t];
    __syncthreads();

    // ---------------- Phase 1: quantize -> big_index (one thread per row) ----
    {
        const int r = rbase + t;
        if (r < B) {
            const v2f* zp = (const v2f*)(z + (size_t)r * NDIM);  // 8B aligned
            v2f a = __builtin_nontemporal_load(zp + 0);          // z read once: NT
            v2f b = __builtin_nontemporal_load(zp + 1);
            v2f c = __builtin_nontemporal_load(zp + 2);
            float zv[NDIM] = {a.x, a.y, b.x, b.y, c.x, c.y};
            int big = 0;
            int basis = 1;
            #pragma unroll
            for (int i = 0; i < NDIM; ++i) {
                int cnt = 0;
                #pragma unroll
                for (int j = 0; j < JLEV; ++j)
                    cnt += (zv[i] >= pshm[j * NDIM + i]) ? 1 : 0;
                int idx = cnt - 1;           // percentiles sorted: max{j: z>=p[j]} = count-1
                if (idx < 0) idx = 0;        // no edge cleared -> 0 (matches reference)
                big += idx * basis;
                basis *= JLEV;
            }
            idx_shm[t] = big;
            __builtin_nontemporal_store((float)big, out_big + r);
        }
    }
    __syncthreads();

    // ---------------- Phase 2: 256B/row codebook gather -----------------------
    // 4096 float4 chunks per block; 16 per thread. chunk q -> row q/16, col q%16.
    constexpr int KCHUNKS = (ROWS * CHAN / 4) / THREADS;   // 16

#if USE_ASYNC
    // Async DMA path: global -> LDS -> global, no VGPR data movement.
    // Loads: default RT so the 64MB codebook stays L2-resident.
    #pragma unroll
    for (int k = 0; k < KCHUNKS; ++k) {
        int q   = t + k * THREADS;
        int row = q >> 4;
        int col = q & 15;
        int gr  = rbase + row;
        if (gr < B) {
            int big = idx_shm[row];
            const float* src = codebook + (size_t)big * CHAN + col * 4;
            __builtin_amdgcn_global_load_async_to_lds_b128(
                (v4i*)src,
                (lds_v4i*)&stage[q * 4],
                0, CPOL_RT);
        }
    }
    __builtin_amdgcn_s_wait_asynccnt(0);   // loads landed in LDS before stores read it
    // Stores: NT — 512MB write-once stream must not evict the codebook from L2.
    #pragma unroll
    for (int k = 0; k < KCHUNKS; ++k) {
        int q   = t + k * THREADS;
        int row = q >> 4;
        int col = q & 15;
        int gr  = rbase + row;
        if (gr < B) {
            float* dst = out_q + (size_t)gr * CHAN + col * 4;
            __builtin_amdgcn_global_store_async_from_lds_b128(
                (v4i*)dst,
                (lds_v4i*)&stage[q * 4],
                0, CPOL_NT);
        }
    }
    // s_endpgm performs an implicit wait-idle, draining ASYNCcnt.
#else
    // Fallback: b128 loads through VGPRs + non-temporal streaming stores.
    (void)stage;
    #pragma unroll
    for (int k = 0; k < KCHUNKS; ++k) {
        int q   = t + k * THREADS;
        int row = q >> 4;
        int col = q & 15;
        int gr  = rbase + row;
        if (gr < B) {
            int big = idx_shm[row];
            const v4f* src = (const v4f*)(codebook + (size_t)big * CHAN + col * 4);
            v4f v = *src;                                      // L2-resident gather
            v4f* dst = (v4f*)(out_q + (size_t)gr * CHAN + col * 4);
            __builtin_nontemporal_store(v, dst);               // don't pollute L2
        }
    }
#endif
}

extern "C" void kernel_launch(void* const* d_in, const int* in_sizes, int n_in,
                              void* d_out, int out_size, void* d_ws, size_t ws_size,
                              hipStream_t stream) {
    (void)n_in; (void)out_size; (void)d_ws; (void)ws_size;
    const float* z        = (const float*)d_in[0];
    const float* perc     = (const float*)d_in[1];
    const float* codebook = (const float*)d_in[2];
    const int B = in_sizes[0] / NDIM;              // 2,000,000
    float* out_q   = (float*)d_out;                // [B, CHAN]
    float* out_big = out_q + (size_t)B * CHAN;     // [B]
    const int blocks = (B + ROWS - 1) / ROWS;
    hipLaunchKernelGGL(fsq_gather_kernel, dim3(blocks), dim3(THREADS), 0, stream,
                       z, perc, codebook, out_q, out_big, B);
}